// RNNModel_55602646614306
// MI455X (gfx1250) — compile-verified
//
#include <hip/hip_runtime.h>
#include <hip/hip_bf16.h>

// ---------------- problem constants ----------------
#define T_SEQ     8192
#define H_DIM     1000
#define N_CHARS   68

// ---------------- recurrence config ----------------
#define NWG         16          // one cluster's worth of workgroups (max 16/cluster)
#define ROWS_MAX    63          // ceil(1000/16) rows of W_hh per WG -> 252 KB LDS slice
#define REC_THREADS 1024        // 32 waves per WG
#define REC_LDS_FLOATS (ROWS_MAX * H_DIM + 1024)   // W slice + padded h vector
#define REC_LDS_BYTES  (REC_LDS_FLOATS * 4)        // 256,096 B <= 320 KB WGP LDS

typedef __attribute__((ext_vector_type(2))) float        v2f;
typedef __attribute__((ext_vector_type(8))) float        v8f;
typedef __attribute__((ext_vector_type(4))) unsigned int u32x4;
typedef __attribute__((ext_vector_type(4))) int          i32x4;
typedef __attribute__((ext_vector_type(8))) int          i32x8;

#if __has_builtin(__builtin_amdgcn_tensor_load_to_lds)
#define HAVE_TDM 1
#else
#define HAVE_TDM 0
#endif

// low 32 bits of a flat shared-aperture pointer == LDS byte address (ISA 10.2)
__device__ __forceinline__ unsigned lds_byte_addr(const void* p) {
    return (unsigned)(size_t)p;
}

// ---------------------------------------------------------------------------
// Kernel 0: init — h_all[0] = initial_h, zero the grid-barrier counter.
// ---------------------------------------------------------------------------
__global__ void k_init(const float* __restrict__ initial_h,
                       float* __restrict__ h_all,
                       unsigned* __restrict__ counter) {
    int i = blockIdx.x * blockDim.x + threadIdx.x;
    if (i < H_DIM) h_all[i] = initial_h[i];
    if (i == 0) *counter = 0u;
}

// ---------------------------------------------------------------------------
// Kernel 1: gather  xh[t][h] = W_xh[h][x_idx[t]] + B_h[h]
// ---------------------------------------------------------------------------
__global__ void k_gather(const int* __restrict__ x_idx,
                         const float* __restrict__ W_xh,
                         const float* __restrict__ B_h,
                         float* __restrict__ xh) {
    const int t = blockIdx.x;
    const int c = x_idx[t];
    for (int h = threadIdx.x; h < H_DIM; h += blockDim.x) {
        xh[(size_t)t * H_DIM + h] = W_xh[(size_t)h * N_CHARS + c] + B_h[h];
    }
}

// ---------------------------------------------------------------------------
// Kernel 2: sequential recurrence (16 persistent WGs = one cluster).
// WG g owns rows [g*63, g*63+nrows) of W_hh, DMA'd once into LDS via the
// Tensor Data Mover (2D tile descriptor, TENSORcnt). Each step:
//   - async-copy h_all[t] (4 KB, L2-resident) into LDS (ASYNCcnt)
//   - 16 lanes per row compute the dot product, shfl-reduce
//   - lane 0 writes tanh(dot + xh[t][row]) to h_all[t+1][row]
//   - cluster barrier (s_barrier_signal/-wait -3) + release/acquire atomic
//     barrier through L2
// ---------------------------------------------------------------------------
__global__ __launch_bounds__(REC_THREADS, 1)
void k_recurrence(const float* __restrict__ W_hh,
                  const float* __restrict__ xh,
                  float* __restrict__ h_all,
                  unsigned* __restrict__ counter) {
    extern __shared__ float smem[];
    float* wsl = smem;                     // ROWS_MAX * H_DIM  weight slice
    float* hsh = smem + ROWS_MAX * H_DIM;  // H_DIM hidden vector (padded)

    const int wg    = blockIdx.x;
    const int row0  = wg * ROWS_MAX;
    const int nrows = min(ROWS_MAX, H_DIM - row0);   // 63, last WG: 55
    const float* __restrict__ wsrc = W_hh + (size_t)row0 * H_DIM;

#if HAVE_TDM
    // ---- TDM: one 2D-tile descriptor moves the whole nrows x 1000 f32 slice
    if (threadIdx.x < 32) {                 // wave 0 issues the DMA
        const unsigned long long ga = (unsigned long long)(size_t)wsrc;
        u32x4 g0;
        g0[0] = 1u;                                   // count=1 (valid user load)
        g0[1] = lds_byte_addr(wsl);                   // lds_addr
        g0[2] = (unsigned)(ga & 0xFFFFFFFFu);         // global_addr[31:0]
        g0[3] = (unsigned)((ga >> 32) & 0x01FFFFFFu)  // global_addr[56:32]
              | (2u << 30);                           // type=2 ("image")
        i32x8 g1;
        g1[0] = (int)(2u << 16);                      // wg_mask=0 | data_size=4B
        g1[1] = (int)((unsigned)H_DIM << 16);         // tensor_dim0[15:0]
        g1[2] = (int)((unsigned)nrows << 16);         // dim0 hi=0 | tensor_dim1 lo
        g1[3] = (int)((unsigned)H_DIM << 16);         // dim1 hi=0 | tile_dim0
        g1[4] = (int)nrows;                           // tile_dim1 | tile_dim2=0
        g1[5] = (int)H_DIM;                           // tensor_dim0_stride[31:0]
        g1[6] = 0;                                    // stride0 hi | stride1 lo
        g1[7] = 0;                                    // stride1 hi
        const i32x4 gz = {0, 0, 0, 0};
#if defined(__clang_major__) && (__clang_major__ >= 23)
        const i32x8 gz8 = {0, 0, 0, 0, 0, 0, 0, 0};
        __builtin_amdgcn_tensor_load_to_lds(g0, g1, gz, gz, gz8, 0);
#else
        __builtin_amdgcn_tensor_load_to_lds(g0, g1, gz, gz, 0);
#endif
        __builtin_amdgcn_s_wait_tensorcnt(0);
    }
#else
    for (int i = threadIdx.x; i < nrows * H_DIM; i += REC_THREADS) {
        __builtin_prefetch(wsrc + i + 4 * REC_THREADS, 0, 0);
        wsl[i] = wsrc[i];
    }
#endif
    __syncthreads();

    const int grp  = threadIdx.x >> 4;    // 64 groups of 16 lanes; group == local row
    const int lane = threadIdx.x & 15;
    const unsigned hsh_base = lds_byte_addr(hsh);

    for (int t = 0; t < T_SEQ; ++t) {
        // ---- async-copy h_all[t] from L2 into LDS (ASYNCcnt path) ----
        {
            const int i = (int)threadIdx.x;
            if (i < H_DIM) {
                const unsigned la = hsh_base + (unsigned)i * 4u;
                const unsigned long long ga =
                    (unsigned long long)(size_t)(h_all + (size_t)t * H_DIM + i);
                asm volatile("global_load_async_to_lds_b32 %0, %1, off"
                             :: "v"(la), "v"(ga) : "memory");
            }
            asm volatile("s_wait_asynccnt 0" ::: "memory");
        }
        __syncthreads();

        // ---- row dot product: 16 lanes per row ----
        if (grp < nrows) {
            const float* __restrict__ wr = wsl + grp * H_DIM;
            float acc = 0.f;
            #pragma unroll 4
            for (int k = lane; k < H_DIM; k += 16)
                acc += wr[k] * hsh[k];
            acc += __shfl_xor(acc, 8, 16);
            acc += __shfl_xor(acc, 4, 16);
            acc += __shfl_xor(acc, 2, 16);
            acc += __shfl_xor(acc, 1, 16);
            if (lane == 0) {
                const int row = row0 + grp;
                float hn = tanhf(acc + xh[(size_t)t * H_DIM + row]);
                h_all[(size_t)(t + 1) * H_DIM + row] = hn;
            }
        }

        // make stores visible at device scope before signaling
        __threadfence();
        __syncthreads();

        // CDNA5 cluster barrier (NOP if dispatch not cluster-formed; the
        // atomic barrier below is the portable guarantee).
#if __has_builtin(__builtin_amdgcn_s_cluster_barrier)
        __builtin_amdgcn_s_cluster_barrier();
#endif
        if (threadIdx.x == 0) {
            __hip_atomic_fetch_add(counter, 1u, __ATOMIC_RELEASE,
                                   __HIP_MEMORY_SCOPE_AGENT);
            const unsigned target = (unsigned)(t + 1) * NWG;
            while (__hip_atomic_load(counter, __ATOMIC_ACQUIRE,
                                     __HIP_MEMORY_SCOPE_AGENT) < target) {
                __builtin_amdgcn_s_sleep(8);
            }
        }
        __syncthreads();
        // acquire in every thread so subsequent h_all loads can't hit stale WGP$
        (void)__hip_atomic_load(counter, __ATOMIC_ACQUIRE,
                                __HIP_MEMORY_SCOPE_AGENT);
    }
}

// ---------------------------------------------------------------------------
// Kernel 3: output projection  out = h_all @ W_hy.T  via V_WMMA_F32_16X16X4_F32
// One wave per 16(M=time) x 16(N=char) tile, K-loop over 1000 in steps of 4.
// Interior tiles (wave-uniform test) run a branch-free, mask-free loop so
// EXEC stays all-1s; boundary tiles neutralize OOB lanes with a 0/1 multiply
// instead of predicated loads (avoids exec-save/restore around every load).
// ---------------------------------------------------------------------------
#define MT_TILES ((T_SEQ + 1 + 15) / 16)   // 513
#define NT_TILES ((N_CHARS + 15) / 16)     // 5

__global__ void k_proj(const float* __restrict__ h_all,
                       const float* __restrict__ W_hy,
                       float* __restrict__ out) {
    const int wid  = (blockIdx.x * blockDim.x + threadIdx.x) >> 5;
    const int lane = threadIdx.x & 31;
    if (wid >= MT_TILES * NT_TILES) return;          // wave-uniform exit

    const int mt = wid / NT_TILES;
    const int nt = wid % NT_TILES;
    const int m0 = mt * 16;
    const int n0 = nt * 16;

    const int half = lane >> 4;       // K sub-pair selector (ISA 7.12.2 layout)
    const int l    = lane & 15;

    const int m = m0 + l;             // A row for this lane
    const int c = n0 + l;             // B col for this lane
    const float* __restrict__ arow =
        h_all + (size_t)(m < T_SEQ + 1 ? m : 0) * H_DIM + half * 2;
    const float* __restrict__ bcol =
        W_hy  + (size_t)(c < N_CHARS ? c : 0) * H_DIM + half * 2;

    const bool interior = (m0 + 16 <= T_SEQ + 1) && (n0 + 16 <= N_CHARS);

    v8f acc = {};
    if (interior) {                                   // 2048 / 2565 tiles
        for (int k = 0; k < H_DIM; k += 4) {          // 250 f32 WMMAs
            v2f a = *(const v2f*)(arow + k);          // 8B-aligned
            v2f b = *(const v2f*)(bcol + k);
            acc = __builtin_amdgcn_wmma_f32_16x16x4_f32(
                false, a, false, b, (short)0, acc, false, false);
        }
    } else {
        const float am = (m < T_SEQ + 1) ? 1.f : 0.f; // branch-free lane masks
        const float bm = (c < N_CHARS)   ? 1.f : 0.f;
        for (int k = 0; k < H_DIM; k += 4) {
            v2f a = *(const v2f*)(arow + k) * am;
            v2f b = *(const v2f*)(bcol + k) * bm;
            acc = __builtin_amdgcn_wmma_f32_16x16x4_f32(
                false, a, false, b, (short)0, acc, false, false);
        }
    }

    // D layout: VGPR r -> M = m0 + r + half*8, N = n0 + l
    #pragma unroll
    for (int r = 0; r < 8; ++r) {
        const int mm = m0 + r + half * 8;
        const int nn = n0 + l;
        if (mm < T_SEQ + 1 && nn < N_CHARS)
            out[(size_t)mm * N_CHARS + nn] = acc[r];
    }
}

// ---------------------------------------------------------------------------
extern "C" void kernel_launch(void* const* d_in, const int* in_sizes, int n_in,
                              void* d_out, int out_size, void* d_ws, size_t ws_size,
                              hipStream_t stream) {
    const int*   x_idx     = (const int*)  d_in[0];
    const float* initial_h = (const float*)d_in[1];
    const float* W_xh      = (const float*)d_in[2];
    const float* W_hh      = (const float*)d_in[3];
    const float* W_hy      = (const float*)d_in[4];
    const float* B_h       = (const float*)d_in[5];
    float*       out       = (float*)d_out;

    // workspace layout (256B aligned regions)
    const size_t xh_bytes   = ((size_t)T_SEQ * H_DIM * sizeof(float) + 255) & ~(size_t)255;
    const size_t hall_bytes = ((size_t)(T_SEQ + 1) * H_DIM * sizeof(float) + 255) & ~(size_t)255;
    char* ws = (char*)d_ws;
    float*    xh      = (float*)ws;
    float*    h_all   = (float*)(ws + xh_bytes);
    unsigned* counter = (unsigned*)(ws + xh_bytes + hall_bytes);

    // allow 256 KB of dynamic LDS for the recurrence kernel (host-side attr,
    // not a stream op — safe under graph capture)
    (void)hipFuncSetAttribute((const void*)k_recurrence,
                              hipFuncAttributeMaxDynamicSharedMemorySize,
                              REC_LDS_BYTES);

    k_init<<<(H_DIM + 255) / 256, 256, 0, stream>>>(initial_h, h_all, counter);
    k_gather<<<T_SEQ, 256, 0, stream>>>(x_idx, W_xh, B_h, xh);
    k_recurrence<<<NWG, REC_THREADS, REC_LDS_BYTES, stream>>>(W_hh, xh, h_all, counter);

    const int total_waves = MT_TILES * NT_TILES;            // 2565
    const int proj_blocks = (total_waves * 32 + 255) / 256; // 321
    k_proj<<<proj_blocks, 256, 0, stream>>>(h_all, W_hy, out);
}